// HashGridEncoder3D_16664473109143
// MI455X (gfx1250) — compile-verified
//
#include <hip/hip_runtime.h>
#include <math.h>

namespace {

constexpr int kL    = 8;
constexpr int kT    = 1 << 18;
constexpr int kMask = kT - 1;
constexpr int kP0   = 1540863;
constexpr int kP1   = 1256879;
constexpr int kP2   = 1957123;

typedef float v4f __attribute__((ext_vector_type(4)));

struct ResParams { float r[kL]; };

__global__ __launch_bounds__(256) void hashgrid_fwd(
    const float* __restrict__ x01,     // [N,3]
    const float* __restrict__ tables,  // [L,T,2]
    float* __restrict__ out,           // [N, L*2]
    ResParams rp, int n)
{
    const int i = blockIdx.x * 256 + (int)threadIdx.x;
    if (i >= n) return;

    float x = x01[3 * (size_t)i + 0];
    float y = x01[3 * (size_t)i + 1];
    float z = x01[3 * (size_t)i + 2];
    x = fminf(fmaxf(x, 0.0f), 1.0f);
    y = fminf(fmaxf(y, 0.0f), 1.0f);
    z = fminf(fmaxf(z, 0.0f), 1.0f);

    float o[2 * kL];

#pragma unroll
    for (int l = 0; l < kL; ++l) {
        const float r  = rp.r[l];
        const float px = x * r, py = y * r, pz = z * r;
        const float flx = floorf(px), fly = floorf(py), flz = floorf(pz);
        const float fx = px - flx, fy = py - fly, fz = pz - flz;
        const int   ix = (int)flx, iy = (int)fly, iz = (int)flz;

        // base hash contributions; "+1" corner is just +prime (int32 exact)
        const int hx0 = ix * kP0, hx1 = hx0 + kP0;
        const int hy0 = iy * kP1, hy1 = hy0 + kP1;
        const int hz0 = iz * kP2, hz1 = hz0 + kP2;

        const float2* __restrict__ tb =
            reinterpret_cast<const float2*>(tables) + (size_t)l * kT;

        // corner k: x = (k>>2)&1, y = (k>>1)&1, z = k&1 (matches OFFSETS order)
        // issue all 8 gathers (global_load_b64) before consuming
        const float2 e0 = tb[(hx0 ^ hy0 ^ hz0) & kMask];
        const float2 e1 = tb[(hx0 ^ hy0 ^ hz1) & kMask];
        const float2 e2 = tb[(hx0 ^ hy1 ^ hz0) & kMask];
        const float2 e3 = tb[(hx0 ^ hy1 ^ hz1) & kMask];
        const float2 e4 = tb[(hx1 ^ hy0 ^ hz0) & kMask];
        const float2 e5 = tb[(hx1 ^ hy0 ^ hz1) & kMask];
        const float2 e6 = tb[(hx1 ^ hy1 ^ hz0) & kMask];
        const float2 e7 = tb[(hx1 ^ hy1 ^ hz1) & kMask];

        const float wx0 = 1.0f - fx, wx1 = fx;
        const float wy0 = 1.0f - fy, wy1 = fy;
        const float wz0 = 1.0f - fz, wz1 = fz;
        // match jnp.prod left-to-right: (wx*wy)*wz
        const float w0 = (wx0 * wy0) * wz0;
        const float w1 = (wx0 * wy0) * wz1;
        const float w2 = (wx0 * wy1) * wz0;
        const float w3 = (wx0 * wy1) * wz1;
        const float w4 = (wx1 * wy0) * wz0;
        const float w5 = (wx1 * wy0) * wz1;
        const float w6 = (wx1 * wy1) * wz0;
        const float w7 = (wx1 * wy1) * wz1;

        float a0 = e0.x * w0;
        float a1 = e0.y * w0;
        a0 = fmaf(e1.x, w1, a0); a1 = fmaf(e1.y, w1, a1);
        a0 = fmaf(e2.x, w2, a0); a1 = fmaf(e2.y, w2, a1);
        a0 = fmaf(e3.x, w3, a0); a1 = fmaf(e3.y, w3, a1);
        a0 = fmaf(e4.x, w4, a0); a1 = fmaf(e4.y, w4, a1);
        a0 = fmaf(e5.x, w5, a0); a1 = fmaf(e5.y, w5, a1);
        a0 = fmaf(e6.x, w6, a0); a1 = fmaf(e6.y, w6, a1);
        a0 = fmaf(e7.x, w7, a0); a1 = fmaf(e7.y, w7, a1);

        o[2 * l + 0] = a0;
        o[2 * l + 1] = a1;
    }

    // 64B row per point, 64B-aligned -> four B128 non-temporal stores
    // (output is write-once; NT keeps the 16MB table resident in L2)
    v4f* __restrict__ dst = reinterpret_cast<v4f*>(out + (size_t)i * (2 * kL));
#pragma unroll
    for (int q = 0; q < 4; ++q) {
        v4f v = { o[4 * q + 0], o[4 * q + 1], o[4 * q + 2], o[4 * q + 3] };
        __builtin_nontemporal_store(v, dst + q);
    }
}

} // namespace

extern "C" void kernel_launch(void* const* d_in, const int* in_sizes, int n_in,
                              void* d_out, int out_size, void* d_ws, size_t ws_size,
                              hipStream_t stream) {
    const float* x01    = (const float*)d_in[0];
    const float* tables = (const float*)d_in[1];
    float*       out    = (float*)d_out;
    const int n = in_sizes[0] / 3;

    // Reproduce Python's RES computation bit-for-bit (double math, then floor)
    ResParams rp;
    const double growth = exp(log(256.0 / 16.0) / 7.0);
    for (int l = 0; l < kL; ++l) {
        rp.r[l] = (float)floor(16.0 * pow(growth, (double)l) + 1e-6);
    }

    dim3 block(256);
    dim3 grid((n + 255) / 256);
    hipLaunchKernelGGL(hashgrid_fwd, grid, block, 0, stream,
                       x01, tables, out, rp, n);
}